// GGNN_12103217840256
// MI455X (gfx1250) — compile-verified
//
#include <hip/hip_runtime.h>
#include <hip/hip_bf16.h>

typedef __attribute__((ext_vector_type(2))) float v2f;
typedef __attribute__((ext_vector_type(8))) float v8f;

#define DD 128           // hidden dim
#define AA 64            // annotation dim
#define BS_STRIDE 144    // LDS row stride in v2f units (bank-conflict-free pad)
#define CHUNK_K 64       // K-chunk staged per barrier (32 k-pairs)
#define CHUNK_PAIRS 32   // CHUNK_K / 2
#define BS_ELEMS (CHUNK_PAIRS * BS_STRIDE)   // 4608 v2f = 36 KB

// ---------------------------------------------------------------------------
// CDNA5 f32 WMMA: D(16x16,f32) = A(16x4,f32) x B(4x16,f32) + C
// ---------------------------------------------------------------------------
__device__ __forceinline__ v8f wmma4(v2f a, v2f b, v8f c) {
  return __builtin_amdgcn_wmma_f32_16x16x4_f32(
      false, a, false, b, (short)0, c, false, false);
}

// Stage CHUNK_K rows of B[128-wide] into LDS, k-pair interleaved:
// Bs[kk2*BS_STRIDE + n] = { B[kc+2*kk2][n], B[kc+2*kk2+1][n] }
__device__ __forceinline__ void stage_chunk(v2f* __restrict__ Bs,
                                            const float* __restrict__ B,
                                            int ldb, int kc) {
  for (int i = threadIdx.x; i < CHUNK_PAIRS * DD; i += 256) {
    const int kk2 = i >> 7, n = i & (DD - 1);
    v2f v;
    v.x = B[(size_t)(kc + 2 * kk2)     * ldb + n];
    v.y = B[(size_t)(kc + 2 * kk2 + 1) * ldb + n];
    Bs[kk2 * BS_STRIDE + n] = v;
  }
}

// Accumulate one CHUNK_K slice into a 16(M) x 128(N) strip.
// A frag: lane -> row m = lane&15, k = kc + k + 2*(lane>>4) + {0,1}  (v2f load)
// B frag: one ds_load_b64 per 16-col tile (pair-interleaved LDS layout).
__device__ __forceinline__ void gemm_chunk(const float* __restrict__ Arow, int lda,
                                           const v2f* __restrict__ Bs, int kc,
                                           int lane, v8f acc[8]) {
  const int m   = lane & 15;
  const int kh  = (lane >> 4) << 1;
  const int col = lane & 15;
  for (int k = 0; k < CHUNK_K; k += 4) {
    v2f a = *(const v2f*)(Arow + (size_t)m * lda + (kc + k + kh));
    const v2f* bp = Bs + ((k + kh) >> 1) * BS_STRIDE + col;
#pragma unroll
    for (int t = 0; t < 8; ++t)
      acc[t] = wmma4(a, bp[t * 16], acc[t]);
  }
}

// Dual-weight version: one A fragment feeds two accumulator sets.
__device__ __forceinline__ void gemm_chunk_dual(const float* __restrict__ Arow, int lda,
                                                const v2f* __restrict__ Bs0,
                                                const v2f* __restrict__ Bs1, int kc,
                                                int lane, v8f acc0[8], v8f acc1[8]) {
  const int m   = lane & 15;
  const int kh  = (lane >> 4) << 1;
  const int col = lane & 15;
  for (int k = 0; k < CHUNK_K; k += 4) {
    v2f a = *(const v2f*)(Arow + (size_t)m * lda + (kc + k + kh));
    const int kro = ((k + kh) >> 1) * BS_STRIDE + col;
    const v2f* bp0 = Bs0 + kro;
    const v2f* bp1 = Bs1 + kro;
#pragma unroll
    for (int t = 0; t < 8; ++t) {
      acc0[t] = wmma4(a, bp0[t * 16], acc0[t]);
      acc1[t] = wmma4(a, bp1[t * 16], acc1[t]);
    }
  }
}

__device__ __forceinline__ float sigm(float v) { return 1.0f / (1.0f + __expf(-v)); }

// ---------------------------------------------------------------------------
// a_buf[n, d] = b_act[d]
// ---------------------------------------------------------------------------
__global__ void k_init_a(float* __restrict__ a, const float* __restrict__ b_act,
                         long long total) {
  long long idx = (long long)blockIdx.x * blockDim.x + threadIdx.x;
  if (idx < total) a[idx] = b_act[idx & (DD - 1)];
}

// ---------------------------------------------------------------------------
// C[l][nrows,128] = A[nrows,128] @ B[l][128,128]   (blockIdx.y = label)
// ---------------------------------------------------------------------------
__global__ __launch_bounds__(256) void k_gemm(const float* __restrict__ Asrc,
                                              const float* __restrict__ Ball,
                                              float* __restrict__ Call,
                                              size_t bStride, size_t cStride,
                                              int nrows) {
  __shared__ v2f Bs[BS_ELEMS];
  const float* B = Ball + blockIdx.y * bStride;
  float* C       = Call + blockIdx.y * cStride;

  const int lane = threadIdx.x & 31;
  const int wave = threadIdx.x >> 5;
  const int row0 = blockIdx.x * 128 + wave * 16;
  const bool active = row0 < nrows;          // wave-uniform

  v8f acc[8];
#pragma unroll
  for (int t = 0; t < 8; ++t) { v8f z = {}; acc[t] = z; }

  for (int kc = 0; kc < DD; kc += CHUNK_K) {
    stage_chunk(Bs, B, DD, kc);
    __syncthreads();
    if (active) gemm_chunk(Asrc + (size_t)row0 * DD, DD, Bs, kc, lane, acc);
    __syncthreads();
  }
  if (!active) return;
  const int rh = (lane >> 4) * 8, col = lane & 15;
#pragma unroll
  for (int t = 0; t < 8; ++t)
#pragma unroll
    for (int j = 0; j < 8; ++j)
      C[(size_t)(row0 + rh + j) * DD + t * 16 + col] = acc[t][j];
}

// ---------------------------------------------------------------------------
// a[dst[e], :] += m[l][src[e], :]   — one float4 gather + 4 atomics per thread
// blockIdx.y = label
// ---------------------------------------------------------------------------
__global__ void k_scatter(const float* __restrict__ m, const int* __restrict__ src,
                          const int* __restrict__ dst, float* __restrict__ a,
                          int E, size_t mStride, long long totalPerLabel) {
  long long idx = (long long)blockIdx.x * blockDim.x + threadIdx.x;
  if (idx >= totalPerLabel) return;
  const int l = blockIdx.y;
  const int e = (int)(idx >> 5);
  const int q = ((int)idx & 31) << 2;
  const int s = src[(size_t)l * E + e];
  const int t = dst[(size_t)l * E + e];
  const float4 v = *(const float4*)(m + l * mStride + (size_t)s * DD + q);
  float* ap = a + (size_t)t * DD + q;
  atomicAdd(ap + 0, v.x);
  atomicAdd(ap + 1, v.y);
  atomicAdd(ap + 2, v.z);
  atomicAdd(ap + 3, v.w);
}

// ---------------------------------------------------------------------------
// z = sigmoid(a@Wz + h@Uz) -> zbuf ; r = sigmoid(a@Wr + h@Ur) ; rhbuf = r*h
// ---------------------------------------------------------------------------
__global__ __launch_bounds__(256) void k_gru1(const float* __restrict__ a,
                                              const float* __restrict__ h,
                                              const float* __restrict__ Wz,
                                              const float* __restrict__ Uz,
                                              const float* __restrict__ Wr,
                                              const float* __restrict__ Ur,
                                              float* __restrict__ zbuf,
                                              float* __restrict__ rhbuf, int nrows) {
  __shared__ v2f Bs0[BS_ELEMS];
  __shared__ v2f Bs1[BS_ELEMS];
  const int lane = threadIdx.x & 31;
  const int wave = threadIdx.x >> 5;
  const int row0 = blockIdx.x * 128 + wave * 16;
  const bool active = row0 < nrows;

  v8f accz[8], accr[8];
#pragma unroll
  for (int t = 0; t < 8; ++t) { v8f z = {}; accz[t] = z; accr[t] = z; }

  for (int kc = 0; kc < DD; kc += CHUNK_K) {           // pass 1: a @ (Wz, Wr)
    stage_chunk(Bs0, Wz, DD, kc);
    stage_chunk(Bs1, Wr, DD, kc);
    __syncthreads();
    if (active)
      gemm_chunk_dual(a + (size_t)row0 * DD, DD, Bs0, Bs1, kc, lane, accz, accr);
    __syncthreads();
  }
  for (int kc = 0; kc < DD; kc += CHUNK_K) {           // pass 2: h @ (Uz, Ur)
    stage_chunk(Bs0, Uz, DD, kc);
    stage_chunk(Bs1, Ur, DD, kc);
    __syncthreads();
    if (active)
      gemm_chunk_dual(h + (size_t)row0 * DD, DD, Bs0, Bs1, kc, lane, accz, accr);
    __syncthreads();
  }
  if (!active) return;
  const int rh = (lane >> 4) * 8, col = lane & 15;
#pragma unroll
  for (int t = 0; t < 8; ++t)
#pragma unroll
    for (int j = 0; j < 8; ++j) {
      const size_t off = (size_t)(row0 + rh + j) * DD + t * 16 + col;
      zbuf[off]  = sigm(accz[t][j]);
      rhbuf[off] = sigm(accr[t][j]) * h[off];
    }
}

// ---------------------------------------------------------------------------
// hcand = tanh(a@W + rh@U) ; hnew = (1-z)*h + z*hcand
// ---------------------------------------------------------------------------
__global__ __launch_bounds__(256) void k_gru2(const float* __restrict__ a,
                                              const float* __restrict__ rhb,
                                              const float* __restrict__ W,
                                              const float* __restrict__ U,
                                              const float* __restrict__ zbuf,
                                              const float* __restrict__ h,
                                              float* __restrict__ hnew, int nrows) {
  __shared__ v2f Bs[BS_ELEMS];
  const int lane = threadIdx.x & 31;
  const int wave = threadIdx.x >> 5;
  const int row0 = blockIdx.x * 128 + wave * 16;
  const bool active = row0 < nrows;

  v8f acc[8];
#pragma unroll
  for (int t = 0; t < 8; ++t) { v8f z = {}; acc[t] = z; }

  for (int kc = 0; kc < DD; kc += CHUNK_K) {           // a @ W
    stage_chunk(Bs, W, DD, kc);
    __syncthreads();
    if (active) gemm_chunk(a + (size_t)row0 * DD, DD, Bs, kc, lane, acc);
    __syncthreads();
  }
  for (int kc = 0; kc < DD; kc += CHUNK_K) {           // rh @ U
    stage_chunk(Bs, U, DD, kc);
    __syncthreads();
    if (active) gemm_chunk(rhb + (size_t)row0 * DD, DD, Bs, kc, lane, acc);
    __syncthreads();
  }
  if (!active) return;
  const int rh = (lane >> 4) * 8, col = lane & 15;
#pragma unroll
  for (int t = 0; t < 8; ++t)
#pragma unroll
    for (int j = 0; j < 8; ++j) {
      const size_t off = (size_t)(row0 + rh + j) * DD + t * 16 + col;
      const float z  = zbuf[off];
      const float hc = tanhf(acc[t][j]);
      hnew[off] = (1.0f - z) * h[off] + z * hc;
    }
}

// ---------------------------------------------------------------------------
// out = [hnew | x] @ rep_kernel + rep_bias   (K = 128 + 64)
// ---------------------------------------------------------------------------
__global__ __launch_bounds__(256) void k_out(const float* __restrict__ hnew,
                                             const float* __restrict__ x,
                                             const float* __restrict__ repK,
                                             const float* __restrict__ repb,
                                             float* __restrict__ out, int nrows) {
  __shared__ v2f Bs[BS_ELEMS];
  const int lane = threadIdx.x & 31;
  const int wave = threadIdx.x >> 5;
  const int row0 = blockIdx.x * 128 + wave * 16;
  const bool active = row0 < nrows;

  v8f acc[8];
#pragma unroll
  for (int t = 0; t < 8; ++t) { v8f z = {}; acc[t] = z; }

  for (int kc = 0; kc < DD; kc += CHUNK_K) {           // hnew @ repK[0:128]
    stage_chunk(Bs, repK, DD, kc);
    __syncthreads();
    if (active) gemm_chunk(hnew + (size_t)row0 * DD, DD, Bs, kc, lane, acc);
    __syncthreads();
  }
  {                                                    // x @ repK[128:192] (K=64)
    stage_chunk(Bs, repK + DD * DD, DD, 0);
    __syncthreads();
    if (active) gemm_chunk(x + (size_t)row0 * AA, AA, Bs, 0, lane, acc);
    __syncthreads();
  }
  if (!active) return;
  const int rh = (lane >> 4) * 8, col = lane & 15;
#pragma unroll
  for (int t = 0; t < 8; ++t)
#pragma unroll
    for (int j = 0; j < 8; ++j)
      out[(size_t)(row0 + rh + j) * DD + t * 16 + col] =
          acc[t][j] + repb[t * 16 + col];
}

// ---------------------------------------------------------------------------
extern "C" void kernel_launch(void* const* d_in, const int* in_sizes, int n_in,
                              void* d_out, int out_size, void* d_ws, size_t ws_size,
                              hipStream_t stream) {
  const float* h     = (const float*)d_in[0];
  const float* x     = (const float*)d_in[1];
  const int*   esrc  = (const int*)d_in[2];
  const int*   edst  = (const int*)d_in[3];
  const float* W_act = (const float*)d_in[4];
  const float* b_act = (const float*)d_in[5];
  const float* Wz    = (const float*)d_in[6];
  const float* Wr    = (const float*)d_in[7];
  const float* W     = (const float*)d_in[8];
  const float* Uz    = (const float*)d_in[9];
  const float* Ur    = (const float*)d_in[10];
  const float* U     = (const float*)d_in[11];
  const float* repK  = (const float*)d_in[12];
  const float* repb  = (const float*)d_in[13];

  const int N = in_sizes[0] / DD;      // 50000 nodes
  const int L = 4;
  const int E = in_sizes[2] / L;       // 400000 edges per label

  const size_t ND = (size_t)N * DD;
  const int gBlocks = (N + 127) / 128;
  const long long totND = (long long)ND;
  const long long totPerLabel = (long long)E * 32;   // float4-quarters per label
  const int sBlocks = (int)((totPerLabel + 255) / 256);

  // Workspace: a | msg(1 or L) | rh | hnew ; z reuses msg (stream-ordered).
  const bool batched = ws_size >= (size_t)(L + 3) * ND * sizeof(float);
  float* a_buf  = (float*)d_ws;
  float* m_buf  = a_buf + ND;
  float* rh_buf = m_buf + (batched ? (size_t)L * ND : ND);
  float* hn_buf = rh_buf + ND;
  float* z_buf  = m_buf;

  k_init_a<<<(int)((totND + 255) / 256), 256, 0, stream>>>(a_buf, b_act, totND);

  if (batched) {
    dim3 gg(gBlocks, L);
    k_gemm<<<gg, 256, 0, stream>>>(h, W_act, m_buf,
                                   (size_t)DD * DD, ND, N);
    dim3 gs(sBlocks, L);
    k_scatter<<<gs, 256, 0, stream>>>(m_buf, esrc, edst, a_buf,
                                      E, ND, totPerLabel);
  } else {
    for (int l = 0; l < L; ++l) {
      k_gemm<<<dim3(gBlocks, 1), 256, 0, stream>>>(
          h, W_act + (size_t)l * DD * DD, m_buf, 0, 0, N);
      k_scatter<<<dim3(sBlocks, 1), 256, 0, stream>>>(
          m_buf, esrc + (size_t)l * E, edst + (size_t)l * E, a_buf,
          E, 0, totPerLabel);
    }
  }

  k_gru1<<<gBlocks, 256, 0, stream>>>(a_buf, h, Wz, Uz, Wr, Ur, z_buf, rh_buf, N);
  k_gru2<<<gBlocks, 256, 0, stream>>>(a_buf, rh_buf, W, U, z_buf, h, hn_buf, N);
  k_out <<<gBlocks, 256, 0, stream>>>(hn_buf, x, repK, repb, (float*)d_out, N);
}